// MultiHeadAttentionQuantum_65481071395644
// MI455X (gfx1250) — compile-verified
//
#include <hip/hip_runtime.h>
#include <math.h>

typedef __attribute__((ext_vector_type(2))) float v2f;
typedef __attribute__((ext_vector_type(8))) float v8f;

#define B_   4
#define S_   1024
#define H_   16
#define DK   4
#define E_   64
#define QCH  8               // query chunks per (b,h) -> 512 blocks total
#define QPB  (S_ / QCH)      // 128 queries per block = 8 waves x 16

// 0.5 (= 1/sqrt(Dk)) * log2(e): folded into the A fragment so the WMMA
// output is directly the argument of exp2 (raw v_exp_f32).
#define HALF_LOG2E 0.7213475204444817f

// ---------------------------------------------------------------------------
// Kernel 1: fused quantum-proj + attention (flash-style, no score spill).
//   proj[s][d] = cos(x[b,s,h*4+d]) * cos(theta[d])  -> LDS (16 KB)
//   scores tile (16x16, K=4) via v_wmma_f32_16x16x4_f32; softmax done in a
//   single pass WITHOUT max-subtraction (|scores| <= 2 -> exp in [e^-2,e^2]).
//   num/den accumulated per-lane, reduced with shfl_xor within 16-lane halves
//   (xor offsets 1,2,4,8 never cross the half-wave boundary, matching the
//   C-matrix row split: lanes 0-15 hold rows 0-7, lanes 16-31 rows 8-15).
// Output: attnout[b][s][h*4+d]  (i.e. [B,S,E] layout, ready for the linear)
// ---------------------------------------------------------------------------
__global__ __launch_bounds__(256) void qattn_kernel(
    const float* __restrict__ x,
    const float* __restrict__ theta,
    float* __restrict__ attnout)
{
    __shared__ float pl[S_ * DK];           // proj, row-major [s][d], 16 KB

    const int bh     = blockIdx.x / QCH;
    const int qchunk = blockIdx.x % QCH;
    const int b      = bh / H_;
    const int h      = bh % H_;

    const float ct0 = __cosf(theta[0]);
    const float ct1 = __cosf(theta[1]);
    const float ct2 = __cosf(theta[2]);
    const float ct3 = __cosf(theta[3]);

    // ---- Phase A: build proj in LDS (each thread: 4 rows, float4 I/O) ----
    {
        const float4* xr  = (const float4*)(x + (size_t)b * S_ * E_ + h * DK);
        float4*       plr = (float4*)pl;
        for (int s = threadIdx.x; s < S_; s += 256) {
            float4 v = xr[s * (E_ / 4)];    // row stride = E_ floats
            float4 p;
            p.x = __cosf(v.x) * ct0;
            p.y = __cosf(v.y) * ct1;
            p.z = __cosf(v.z) * ct2;
            p.w = __cosf(v.w) * ct3;
            plr[s] = p;
        }
    }
    __syncthreads();

    // ---- Phase B: one 16-query tile per wave ----
    const int wave = threadIdx.x >> 5;
    const int lane = threadIdx.x & 31;
    const int mn   = lane & 15;             // M (A) / N (B) index within tile
    const bool hi  = (lane >= 16);          // upper half-wave -> K = 2,3

    const int qb = qchunk * QPB + wave * 16;

    // A fragment: A[m][k] = proj[qb+m][k] * 0.5*log2(e)
    // (A-matrix 16x4 f32 layout: lane l -> M=l&15, K=(l>=16?2:0)+{0,1})
    v2f a;
    {
        float4 q = ((const float4*)pl)[qb + mn];
        a.x = (hi ? q.z : q.x) * HALF_LOG2E;
        a.y = (hi ? q.w : q.y) * HALF_LOG2E;
    }

    float num[8][4];
    float den[8];
#pragma unroll
    for (int r = 0; r < 8; ++r) {
        den[r] = 0.0f;
        num[r][0] = 0.0f; num[r][1] = 0.0f; num[r][2] = 0.0f; num[r][3] = 0.0f;
    }

    for (int kt = 0; kt < S_; kt += 16) {
        // key-tile row for this lane (lanes l and l+16 read same addr: bcast)
        float4 kv = ((const float4*)pl)[kt + mn];
        v2f bf;
        bf.x = hi ? kv.z : kv.x;            // B[k][n] = proj[kt+n][k]
        bf.y = hi ? kv.w : kv.y;

        v8f c = {};
        c = __builtin_amdgcn_wmma_f32_16x16x4_f32(
                /*neg_a=*/false, a, /*neg_b=*/false, bf,
                /*c_mod=*/(short)0, c, /*reuse_a=*/false, /*reuse_b=*/false);

        // c[r] = log2e * 0.5 * (q.k) for row (hi?8:0)+r, column kt+mn.
        // exp(0.5*qk) == exp2(c[r]); |c| <= 2.9 so raw v_exp_f32 is exact-safe.
#pragma unroll
        for (int r = 0; r < 8; ++r) {
            float e = __builtin_amdgcn_exp2f(c[r]);
            den[r]    += e;
            num[r][0] += e * kv.x;
            num[r][1] += e * kv.y;
            num[r][2] += e * kv.z;
            num[r][3] += e * kv.w;
        }
    }

    // Row reduction across the 16 columns of each half-wave.
#pragma unroll
    for (int off = 1; off < 16; off <<= 1) {
#pragma unroll
        for (int r = 0; r < 8; ++r) {
            den[r]    += __shfl_xor(den[r],    off, 32);
            num[r][0] += __shfl_xor(num[r][0], off, 32);
            num[r][1] += __shfl_xor(num[r][1], off, 32);
            num[r][2] += __shfl_xor(num[r][2], off, 32);
            num[r][3] += __shfl_xor(num[r][3], off, 32);
        }
    }

    // Lanes p<8 of each half write row (hi?8:0)+p.
    const int p = mn;
    if (p < 8) {
        const int  row = qb + (hi ? 8 : 0) + p;
        const float inv = 1.0f / den[p];
        float* o = attnout + ((size_t)(b * S_ + row) * E_ + h * DK);
        float4 v;
        v.x = num[p][0] * inv;
        v.y = num[p][1] * inv;
        v.z = num[p][2] * inv;
        v.w = num[p][3] * inv;
        *(float4*)o = v;
    }
}

// ---------------------------------------------------------------------------
// Kernel 2: out = attnout @ W^T + bias   ([4096,64] x [64,64] GEMM)
// Chained f32 WMMAs, K=64 in 16 steps. One 16x16 output tile per wave.
// grid = 256 blocks (M tiles), block = 128 threads (4 waves = 4 N tiles).
// W is 16 KB -> L2 resident after first touch; no LDS staging needed.
// ---------------------------------------------------------------------------
__global__ __launch_bounds__(128) void linear_kernel(
    const float* __restrict__ attnout,
    const float* __restrict__ W,
    const float* __restrict__ bias,
    float* __restrict__ out)
{
    const int wave = threadIdx.x >> 5;
    const int lane = threadIdx.x & 31;
    const int mn   = lane & 15;
    const int kb   = (lane >> 4) << 1;      // 0 or 2

    const int mb = blockIdx.x * 16;
    const int nb = wave * 16;

    const float* arow = attnout + (size_t)(mb + mn) * E_;   // A[m][*]
    const float* wrow = W       + (size_t)(nb + mn) * E_;   // B[k][n] = W[n][k]

    v8f c = {};
#pragma unroll
    for (int k = 0; k < E_; k += 4) {
        v2f a, bf;
        a.x  = arow[k + kb];
        a.y  = arow[k + kb + 1];
        bf.x = wrow[k + kb];
        bf.y = wrow[k + kb + 1];
        c = __builtin_amdgcn_wmma_f32_16x16x4_f32(
                false, a, false, bf, (short)0, c, false, false);
    }

    const float bv  = bias[nb + mn];
    const int row0  = mb + ((lane >= 16) ? 8 : 0);
#pragma unroll
    for (int r = 0; r < 8; ++r) {
        out[(size_t)(row0 + r) * E_ + nb + mn] = c[r] + bv;
    }
}

// ---------------------------------------------------------------------------
extern "C" void kernel_launch(void* const* d_in, const int* in_sizes, int n_in,
                              void* d_out, int out_size, void* d_ws, size_t ws_size,
                              hipStream_t stream)
{
    const float* x     = (const float*)d_in[0];   // [4,1024,64]
    const float* theta = (const float*)d_in[1];   // [4]
    const float* W     = (const float*)d_in[2];   // [64,64]
    const float* bias  = (const float*)d_in[3];   // [64]
    float*       out   = (float*)d_out;           // [4,1024,64]
    float*       attn  = (float*)d_ws;            // [4,1024,64] scratch = 1 MB

    (void)in_sizes; (void)n_in; (void)out_size; (void)ws_size;

    qattn_kernel<<<B_ * H_ * QCH, 256, 0, stream>>>(x, theta, attn);
    linear_kernel<<<(B_ * S_) / 16, 128, 0, stream>>>(attn, W, bias, out);
}